// Atomic_node_only_lstm_48928267436032
// MI455X (gfx1250) — compile-verified
//
#include <hip/hip_runtime.h>
#include <hip/hip_bf16.h>

// Sizes from the reference
#define NTOT 65536
#define SQ   5
#define KN   4
#define DD   12
#define HH   48

#define ELEMS_PER_BLOCK 128   // 8 waves * 16-row batch tiles
#define THREADS 256
#define WSTR    50            // padded LDS row stride for staged weights

typedef float v2f __attribute__((ext_vector_type(2)));
typedef float v8f __attribute__((ext_vector_type(8)));

// ---- address-space typed accessors (force global_load_* / ds_* codegen) ----
typedef const __attribute__((address_space(1))) float* gcfp;
typedef const __attribute__((address_space(1))) v2f*   gv2p;
typedef __attribute__((address_space(1))) float*       gfp;
typedef const __attribute__((address_space(3))) float* lcfp;
typedef const __attribute__((address_space(3))) v2f*   lv2p;
typedef __attribute__((address_space(3))) float*       lfp;

static __device__ __forceinline__ float ldg1(const float* p) { return *(gcfp)p; }
static __device__ __forceinline__ v2f   ldg2(const float* p) { return *(gv2p)p; }
static __device__ __forceinline__ void  stg1(float* p, float v) { *(gfp)p = v; }
static __device__ __forceinline__ v2f   ldl2(const float* p) { return *(lv2p)p; }
static __device__ __forceinline__ void  stl1(float* p, float v) { *(lfp)p = v; }

static __device__ __forceinline__ float sigf(float x) {
    return 1.0f / (1.0f + __expf(-x));
}
static __device__ __forceinline__ v8f splat8(float x) {
    return (v8f){x, x, x, x, x, x, x, x};
}

// D = A(16x4 f32) * B(4x16 f32) + C(16x16 f32), wave32
static __device__ __forceinline__ v8f wmma4(v2f a, v2f b, v8f c) {
    return __builtin_amdgcn_wmma_f32_16x16x4_f32(
        false, a, false, b, (short)0, c, false, false);
}

// Cooperative copy of a [192][48] weight matrix into LDS with padded stride.
static __device__ __forceinline__ void stage_weights(float* dst, const float* src,
                                                     int tid) {
#pragma unroll 1
    for (int i = tid; i < 192 * HH; i += THREADS) {
        const int r = i / HH, c = i - r * HH;
        stl1(dst + r * WSTR + c, ldg1(src + i));
    }
}

// One LSTM direction over 5 timesteps for a 16-row batch tile.
// Weights already staged in LDS (row stride WSTR).
static __device__ __forceinline__ void lstm_dir(
    const float* xw,            // LDS x for this tile, row stride 240
    float* hs,                  // LDS per-wave [16][48] recurrence bounce
    float* hn, int colOff,      // LDS per-wave [16][96] final h output
    const float* lwih, const float* lwhh,   // LDS staged weights
    const float* bih, const float* bhh,     // global biases
    int laneN, int hi, int reverse) {
    for (int i = laneN + 16 * hi; i < 16 * HH; i += 32) stl1(hs + i, 0.0f);

    float bb[12];
#pragma unroll
    for (int n = 0; n < 12; ++n)
        bb[n] = ldg1(bih + n * 16 + laneN) + ldg1(bhh + n * 16 + laneN);

    v8f cst[3];
    cst[0] = splat8(0.0f); cst[1] = splat8(0.0f); cst[2] = splat8(0.0f);

#pragma unroll 1
    for (int st = 0; st < SQ; ++st) {
        const int t = reverse ? (SQ - 1 - st) : st;

        v8f acc[12];
#pragma unroll
        for (int n = 0; n < 12; ++n) acc[n] = splat8(bb[n]);

        // x_t @ w_ih^T  (K = 48): A + 12 B ds loads (clause), 12-wmma burst
        const float* xr = xw + laneN * (SQ * HH) + t * HH;
        const float* wi = lwih + laneN * WSTR + 2 * hi;
#pragma unroll 1
        for (int k0 = 0; k0 < HH; k0 += 4) {
            v2f a = ldl2(xr + k0 + 2 * hi);
            v2f b[12];
#pragma unroll
            for (int n = 0; n < 12; ++n)
                b[n] = ldl2(wi + n * 16 * WSTR + k0);
#pragma unroll
            for (int n = 0; n < 12; ++n) acc[n] = wmma4(a, b[n], acc[n]);
        }
        // h_{t-1} @ w_hh^T  (K = 48)
        const float* hr = hs + laneN * HH;
        const float* wh = lwhh + laneN * WSTR + 2 * hi;
#pragma unroll 1
        for (int k0 = 0; k0 < HH; k0 += 4) {
            v2f a = ldl2(hr + k0 + 2 * hi);
            v2f b[12];
#pragma unroll
            for (int n = 0; n < 12; ++n)
                b[n] = ldl2(wh + n * 16 * WSTR + k0);
#pragma unroll
            for (int n = 0; n < 12; ++n) acc[n] = wmma4(a, b[n], acc[n]);
        }

        // gates: tiles 0-2 = i, 3-5 = f, 6-8 = g, 9-11 = o
#pragma unroll
        for (int j = 0; j < 3; ++j) {
#pragma unroll
            for (int r = 0; r < 8; ++r) {
                float ci = sigf(acc[3 + j][r]) * cst[j][r] +
                           sigf(acc[j][r]) * tanhf(acc[6 + j][r]);
                cst[j][r] = ci;
                float hv = sigf(acc[9 + j][r]) * tanhf(ci);
                const int row = r + 8 * hi;
                stl1(hs + row * HH + j * 16 + laneN, hv);
                stl1(hn + row * 96 + colOff + j * 16 + laneN, hv);
            }
        }
    }
}

__global__ __launch_bounds__(THREADS)
void fused_gnn_lstm_kernel(
    const float* __restrict__ nodes_feature, const float* __restrict__ pos,
    const float* __restrict__ attmat,
    const float* __restrict__ msg_w, const float* __restrict__ msg_b,
    const float* __restrict__ gwih, const float* __restrict__ gwhh,
    const float* __restrict__ gbih, const float* __restrict__ gbhh,
    const float* __restrict__ wih_fw, const float* __restrict__ whh_fw,
    const float* __restrict__ bih_fw, const float* __restrict__ bhh_fw,
    const float* __restrict__ wih_bw, const float* __restrict__ whh_bw,
    const float* __restrict__ bih_bw, const float* __restrict__ bhh_bw,
    const float* __restrict__ en1_w, const float* __restrict__ en1_b,
    const float* __restrict__ en2_w, const float* __restrict__ en2_b,
    const float* __restrict__ en3_w, const float* __restrict__ en3_b,
    float* __restrict__ out) {
    extern __shared__ float smem[];
    float* xbuf = smem;                          // [128][240]
    float* hsb  = smem + ELEMS_PER_BLOCK * 240;  // [128][48]
    float* hnb  = hsb + ELEMS_PER_BLOCK * HH;    // [128][96]
    float* wlih = hnb + ELEMS_PER_BLOCK * 96;    // [192][WSTR] staged w_ih
    float* wlhh = wlih + 192 * WSTR;             // [192][WSTR] staged w_hh

    const int tid = threadIdx.x;

    if (tid == 0) {
        __builtin_prefetch(wih_fw, 0, 1);
        __builtin_prefetch(whh_fw, 0, 1);
        __builtin_prefetch(wih_bw, 0, 1);
        __builtin_prefetch(whh_bw, 0, 1);
    }

    // ---------------- Phase 1: GRU message passing (scalar, 1 lane = 1 elem) --
    if (tid < ELEMS_PER_BLOCK) {
        const int e = blockIdx.x * ELEMS_PER_BLOCK + tid;
#pragma unroll 1
        for (int s = 0; s < SQ; ++s) {
            float h[KN][DD];
#pragma unroll
            for (int k = 0; k < KN; ++k) {
                const int base = ((e * SQ + s) * KN + k) * 6;
#pragma unroll
                for (int c = 0; c < 6; ++c) {
                    h[k][c]     = ldg1(nodes_feature + base + c);
                    h[k][6 + c] = ldg1(pos + base + c);
                }
            }
#pragma unroll 1
            for (int pass = 0; pass < 2; ++pass) {
                float m_all[KN][DD];
#pragma unroll
                for (int w = 0; w < KN; ++w)
#pragma unroll
                    for (int d = 0; d < DD; ++d) {
                        float acc = msg_b[d];
#pragma unroll
                        for (int j = 0; j < DD; ++j)
                            acc += h[w][j] * msg_w[d * DD + j];
                        m_all[w][d] = acc;
                    }
#pragma unroll 1
                for (int n = 0; n < KN; ++n) {
                    float at[KN];
#pragma unroll
                    for (int w = 0; w < KN; ++w)
                        at[w] = ldg1(attmat + ((e * SQ + s) * KN + n) * KN + w);
                    float mv[DD];
#pragma unroll
                    for (int d = 0; d < DD; ++d)
                        mv[d] = at[0] * m_all[0][d] + at[1] * m_all[1][d] +
                                at[2] * m_all[2][d] + at[3] * m_all[3][d];
                    float tmp[DD];
#pragma unroll
                    for (int d = 0; d < DD; ++d) {
                        float gir = gbih[d],        ghr = gbhh[d];
                        float giz = gbih[DD + d],   ghz = gbhh[DD + d];
                        float gin = gbih[2*DD + d], ghn = gbhh[2*DD + d];
#pragma unroll
                        for (int j = 0; j < DD; ++j) {
                            gir += mv[j]   * gwih[d * DD + j];
                            ghr += h[n][j] * gwhh[d * DD + j];
                            giz += mv[j]   * gwih[(DD + d) * DD + j];
                            ghz += h[n][j] * gwhh[(DD + d) * DD + j];
                            gin += mv[j]   * gwih[(2*DD + d) * DD + j];
                            ghn += h[n][j] * gwhh[(2*DD + d) * DD + j];
                        }
                        const float rr = sigf(gir + ghr);
                        const float zz = sigf(giz + ghz);
                        const float nn = tanhf(gin + rr * ghn);
                        tmp[d] = (1.0f - zz) * nn + zz * h[n][d];
                    }
#pragma unroll
                    for (int d = 0; d < DD; ++d) h[n][d] = tmp[d];
                }
            }
#pragma unroll
            for (int k = 0; k < KN; ++k)
#pragma unroll
                for (int d = 0; d < DD; ++d)
                    stl1(xbuf + tid * (SQ * HH) + s * HH + d * KN + k, h[k][d]);
        }
    }
    __syncthreads();

    // ---------------- Phase 2: WMMA LSTM + encoder, 1 wave = 16 batch rows ----
    const int wid   = tid >> 5;
    const int lane  = tid & 31;
    const int laneN = lane & 15;
    const int hi    = lane >> 4;

    const float* xw = xbuf + (wid * 16) * (SQ * HH);
    float* hs = hsb + wid * 16 * HH;
    float* hn = hnb + wid * 16 * 96;

    // forward direction: stage weights, run
    stage_weights(wlih, wih_fw, tid);
    stage_weights(wlhh, whh_fw, tid);
    __syncthreads();
    lstm_dir(xw, hs, hn, 0, wlih, wlhh, bih_fw, bhh_fw, laneN, hi, 0);
    __syncthreads();
    // backward direction: restage, run
    stage_weights(wlih, wih_bw, tid);
    stage_weights(wlhh, whh_bw, tid);
    __syncthreads();
    lstm_dir(xw, hs, hn, HH, wlih, wlhh, bih_bw, bhh_bw, laneN, hi, 1);

    // ---- encoder layer 1: [16,96] @ en1_w^T -> [16,48], relu
    v8f e1[3];
#pragma unroll
    for (int n = 0; n < 3; ++n) e1[n] = splat8(ldg1(en1_b + n * 16 + laneN));
    {
        const float* ar = hn + laneN * 96;
#pragma unroll 1
        for (int k0 = 0; k0 < 96; k0 += 4) {
            v2f a = ldl2(ar + k0 + 2 * hi);
            v2f b[3];
#pragma unroll
            for (int n = 0; n < 3; ++n)
                b[n] = ldg2(en1_w + (n * 16 + laneN) * 96 + k0 + 2 * hi);
#pragma unroll
            for (int n = 0; n < 3; ++n) e1[n] = wmma4(a, b[n], e1[n]);
        }
    }
#pragma unroll
    for (int n = 0; n < 3; ++n)
#pragma unroll
        for (int r = 0; r < 8; ++r)
            stl1(hs + (r + 8 * hi) * HH + n * 16 + laneN, fmaxf(e1[n][r], 0.0f));

    // ---- encoder layer 2: [16,48] @ en2_w^T -> [16,36], relu (pad N to 48)
    v8f e2[3];
#pragma unroll
    for (int n = 0; n < 3; ++n) {
        int col = n * 16 + laneN; if (col > 35) col = 35;
        e2[n] = splat8(ldg1(en2_b + col));
    }
    {
        const float* ar = hs + laneN * HH;
#pragma unroll 1
        for (int k0 = 0; k0 < HH; k0 += 4) {
            v2f a = ldl2(ar + k0 + 2 * hi);
            v2f b[3];
#pragma unroll
            for (int n = 0; n < 3; ++n) {
                int row = n * 16 + laneN; if (row > 35) row = 35;
                b[n] = ldg2(en2_w + row * HH + k0 + 2 * hi);
            }
#pragma unroll
            for (int n = 0; n < 3; ++n) e2[n] = wmma4(a, b[n], e2[n]);
        }
    }
#pragma unroll
    for (int n = 0; n < 3; ++n)
#pragma unroll
        for (int r = 0; r < 8; ++r) {
            const int col = n * 16 + laneN;
            if (col < 36)
                stl1(hn + (r + 8 * hi) * 96 + col, fmaxf(e2[n][r], 0.0f));
        }

    // ---- encoder layer 3: [16,36] @ en3_w^T -> [16,6]
    int r3 = laneN; if (r3 > 5) r3 = 5;
    v8f e3 = splat8(ldg1(en3_b + r3));
    {
        const float* ar = hn + laneN * 96;
#pragma unroll 1
        for (int k0 = 0; k0 < 36; k0 += 4) {
            v2f a = ldl2(ar + k0 + 2 * hi);
            v2f b = ldg2(en3_w + r3 * 36 + k0 + 2 * hi);
            e3 = wmma4(a, b, e3);
        }
    }
    const int gbase = blockIdx.x * ELEMS_PER_BLOCK + wid * 16;
#pragma unroll
    for (int r = 0; r < 8; ++r) {
        const int row = r + 8 * hi;
        if (laneN < 6) stg1(out + (gbase + row) * 6 + laneN, e3[r]);
    }
}

extern "C" void kernel_launch(void* const* d_in, const int* in_sizes, int n_in,
                              void* d_out, int out_size, void* d_ws, size_t ws_size,
                              hipStream_t stream) {
    (void)in_sizes; (void)n_in; (void)out_size; (void)d_ws; (void)ws_size;
    const float* nodes_feature = (const float*)d_in[0];
    const float* pos           = (const float*)d_in[1];
    const float* attmat        = (const float*)d_in[2];
    const float* msg_w  = (const float*)d_in[3];
    const float* msg_b  = (const float*)d_in[4];
    const float* gwih   = (const float*)d_in[5];
    const float* gwhh   = (const float*)d_in[6];
    const float* gbih   = (const float*)d_in[7];
    const float* gbhh   = (const float*)d_in[8];
    const float* wih_fw = (const float*)d_in[9];
    const float* whh_fw = (const float*)d_in[10];
    const float* bih_fw = (const float*)d_in[11];
    const float* bhh_fw = (const float*)d_in[12];
    const float* wih_bw = (const float*)d_in[13];
    const float* whh_bw = (const float*)d_in[14];
    const float* bih_bw = (const float*)d_in[15];
    const float* bhh_bw = (const float*)d_in[16];
    const float* en1_w = (const float*)d_in[17];
    const float* en1_b = (const float*)d_in[18];
    const float* en2_w = (const float*)d_in[19];
    const float* en2_b = (const float*)d_in[20];
    const float* en3_w = (const float*)d_in[21];
    const float* en3_b = (const float*)d_in[22];
    float* out = (float*)d_out;

    const size_t shmem = (size_t)(ELEMS_PER_BLOCK * 240 +   // xbuf
                                  ELEMS_PER_BLOCK * HH +    // h bounce
                                  ELEMS_PER_BLOCK * 96 +    // hn
                                  2 * 192 * WSTR)           // staged weights
                         * sizeof(float);
    dim3 grid(NTOT / ELEMS_PER_BLOCK), block(THREADS);
    fused_gnn_lstm_kernel<<<grid, block, shmem, stream>>>(
        nodes_feature, pos, attmat, msg_w, msg_b,
        gwih, gwhh, gbih, gbhh,
        wih_fw, whh_fw, bih_fw, bhh_fw,
        wih_bw, whh_bw, bih_bw, bhh_bw,
        en1_w, en1_b, en2_w, en2_b, en3_w, en3_b, out);
}